// VarianceKPNet_86045374808369
// MI455X (gfx1250) — compile-verified
//
#include <hip/hip_runtime.h>
#include <stdint.h>

// ---------------------------------------------------------------------------
// VarianceKPNet pipeline for MI455X (gfx1250, wave32).
//
// d_in[0] = reliability [8,1,256,256] f32
// d_in[1] = variance    [8,1,256,256] f32
// d_in[2] = desc        [8,64,256,256] f32
// d_in[3] = top_k (int, ==4096)
// d_out   = [ mk_top 8*4096*2 | sc_top 8*4096 | feats 8*4096*64 ] f32
//
// ws layout: [0..8)   minmax (2 x u32, float-bit encoded, values >= 0)
//            [256..)  svar    8*65536 f32  (var_s if NMS mask else -1)
//            [+2MB..) invnorm 8*65536 f32  (1/max(||desc||_c,1e-12) per pixel)
//            [+4MB..) keys    8*65536 u64  (sort keys)
// ---------------------------------------------------------------------------

#define Bv 8
#define Hm 256
#define Wm 256
#define NP 65536            // Hm*Wm
#define CC 64
#define TOPK 4096

typedef __attribute__((ext_vector_type(2))) float v2f;
typedef __attribute__((ext_vector_type(8))) float v8f;
typedef __attribute__((ext_vector_type(4))) unsigned int v4u;
typedef __attribute__((ext_vector_type(8))) int v8i;
typedef __attribute__((ext_vector_type(4))) int v4i;

// ---- helpers --------------------------------------------------------------

__device__ __forceinline__ float unnorm_q(int q) {
  // _unnorm(4*q, 1024, 256) = (4q/1023)*256 - 0.5
  return (4.0f * (float)q / 1023.0f) * 256.0f - 0.5f;
}

__device__ __forceinline__ float bilin_var(const float* __restrict__ var,
                                           int b, int px, int py) {
  float ix = unnorm_q(px), iy = unnorm_q(py);
  float fx0 = floorf(ix), fy0 = floorf(iy);
  int x0 = (int)fx0, y0 = (int)fy0;
  float tx = ix - fx0, ty = iy - fy0;
  const float* m = var + ((size_t)b << 16);
  float v00 = (x0   >= 0 && x0   < Wm && y0   >= 0 && y0   < Hm) ? m[y0*Wm + x0]       : 0.0f;
  float v01 = (x0+1 >= 0 && x0+1 < Wm && y0   >= 0 && y0   < Hm) ? m[y0*Wm + x0+1]     : 0.0f;
  float v10 = (x0   >= 0 && x0   < Wm && y0+1 >= 0 && y0+1 < Hm) ? m[(y0+1)*Wm + x0]   : 0.0f;
  float v11 = (x0+1 >= 0 && x0+1 < Wm && y0+1 >= 0 && y0+1 < Hm) ? m[(y0+1)*Wm + x0+1] : 0.0f;
  return (v00*(1.0f-tx) + v01*tx)*(1.0f-ty) + (v10*(1.0f-tx) + v11*tx)*ty;
}

__device__ __forceinline__ float cub1(float x) {  // A=-0.75
  return ((-0.75f + 2.0f)*x - (-0.75f + 3.0f))*x*x + 1.0f;
}
__device__ __forceinline__ float cub2(float x) {
  return ((-0.75f*x - 5.0f*-0.75f)*x + 8.0f*-0.75f)*x - 4.0f*-0.75f;
}
__device__ __forceinline__ void cubw(float t, float w[4]) {
  w[0] = cub2(t + 1.0f); w[1] = cub1(t); w[2] = cub1(1.0f - t); w[3] = cub2(2.0f - t);
}

// ---- kernel 0: init workspace scalars -------------------------------------

__global__ void vkp_init(unsigned int* mm) {
  mm[0] = 0x7F800000u;  // +inf  (min slot; all var_s >= 0 so uint order == float order)
  mm[1] = 0x00000000u;  // 0     (max slot)
}

// ---- kernel 1: NMS + bilinear variance (TDM strip load into LDS) ----------
// grid: 8 batches x 16 strips of 16 rows. block = 256.

__global__ void vkp_nms(const float* __restrict__ rel,
                        const float* __restrict__ var,
                        float* __restrict__ svar,
                        unsigned int* __restrict__ mm) {
  __shared__ float s[20 * Wm];  // rows y0-2 .. y0+17, zero-padded at image edges
  int b = blockIdx.x >> 4;
  int strip = blockIdx.x & 15;
  int y0 = strip * 16;
  int t = threadIdx.x;

  int gy0 = y0 - 2;
  int start = gy0 < 0 ? 0 : gy0;
  int end = (gy0 + 20 > Hm) ? Hm : (gy0 + 20);
  int rows_t = end - start;
  int wr = start - gy0;  // first LDS row written by the TDM

  // Zero LDS halo rows not covered by the clamped tile (image boundary).
  for (int i = t; i < 20 * Wm; i += 256) {
    int r = i >> 8;
    if (r < wr || r >= wr + rows_t) s[i] = 0.0f;
  }

  // Wave 0 issues one Tensor-Data-Mover load: (rows_t x 256) f32 tile,
  // global stride 256 elems, into LDS at row `wr`.
  if (t < 32) {
    unsigned long long ga =
        (unsigned long long)(uintptr_t)(rel + ((size_t)b << 16) + (size_t)start * Wm);
    unsigned int lds_off = (unsigned int)(uintptr_t)(&s[wr * Wm]);
    v4u g0;
    g0.x = 1u;                                        // count=1 (valid user D#)
    g0.y = lds_off;                                   // lds_addr [63:32]
    g0.z = (unsigned int)(ga & 0xFFFFFFFFull);        // global_addr lo
    g0.w = (unsigned int)((ga >> 32) & 0x01FFFFFFull) // global_addr hi (57-bit)
           | 0x80000000u;                             // type=2 ("image") in [127:126]
    v8i g1;
    g1[0] = 0x00020000;                 // data_size=2 (4 bytes); no multicast/pad/iter
    g1[1] = (int)(256u << 16);          // tensor_dim0 = 256 (low16 at bits 63:48)
    g1[2] = 0;                          // tensor_dim0 hi16 = 0; tensor_dim1 lo16 = 0
    g1[3] = (int)(1u | (256u << 16));   // tensor_dim1 = 0x10000; tile_dim0 = 256
    g1[4] = rows_t;                     // tile_dim1 = rows; tile_dim2 = 0
    g1[5] = 256;                        // tensor_dim0_stride (elems), low 32
    g1[6] = 0;                          // stride0 hi16 = 0; stride1 lo16 = 0
    g1[7] = 0;
    v4i gz = {0, 0, 0, 0};              // 2D tensor: groups 2/3 unused
    v8i gz8 = {0, 0, 0, 0, 0, 0, 0, 0}; // extra group (6-arg toolchain), unused
    __builtin_amdgcn_tensor_load_to_lds(g0, g1, gz, gz, gz8, 0);
    __builtin_amdgcn_s_wait_tensorcnt(0);
  }
  __syncthreads();

  // 5x5 NMS (zero padding == -inf padding since rel >= 0) + variance sample.
  for (int rr = 0; rr < 16; rr++) {
    int x = t;
    int y = y0 + rr;
    float v = s[(rr + 2) * Wm + x];
    float lm = v;
#pragma unroll
    for (int dy = -2; dy <= 2; dy++) {
      const float* row = &s[(rr + 2 + dy) * Wm];
#pragma unroll
      for (int dx = -2; dx <= 2; dx++) {
        int xx = x + dx;
        if (xx >= 0 && xx < Wm) lm = fmaxf(lm, row[xx]);
      }
    }
    bool msk = (v == lm) && (v > 0.05f);
    int p = y * Wm + x;
    float outv = -1.0f;
    if (msk || p == 0) {
      // Reference takes min/max over ALL N entries: keypoints + the padding
      // entries, which all bilinear-sample at mk=(0,0) == pixel (0,0)'s value.
      float vs = bilin_var(var, b, x, y);
      if (msk) outv = vs;
      atomicMin(&mm[0], __float_as_uint(vs));
      atomicMax(&mm[1], __float_as_uint(vs));
    }
    svar[((size_t)b << 16) + p] = outv;
  }
}

// ---- kernel 2: score + pack sort keys -------------------------------------

__global__ void vkp_score(const float* __restrict__ rel,
                          const float* __restrict__ svar,
                          const unsigned int* __restrict__ mm,
                          unsigned long long* __restrict__ keys) {
  int tid = blockIdx.x * 256 + threadIdx.x;   // [0, 8*65536)
  int b = tid >> 16, p = tid & (NP - 1);
  float sv = svar[tid];
  float score = -1.0f;
  if (sv >= 0.0f && p != 0) {                 // kp==(0,0) forced to -1 by reference
    int x = p & 255, y = p >> 8;
    int xi = (int)rintf(unnorm_q(x));         // round-half-even == jnp.round
    int yi = (int)rintf(unnorm_q(y));
    float r = (xi >= 0 && xi < Wm && yi >= 0 && yi < Hm)
                  ? rel[((size_t)b << 16) + yi * Wm + xi] : 0.0f;
    float mn = __uint_as_float(mm[0]);
    float mx = __uint_as_float(mm[1]);
    float vn = (sv - mn) / (mx - mn + 1e-6f);
    score = r * (1.0f - vn);
  }
  // Monotone flip: ascending u64 key  <=>  (score desc, pixel asc) == stable
  // argsort(-scores) of the row-major-compacted list.
  unsigned int u = __float_as_uint(score);
  unsigned int mfl = (u & 0x80000000u) ? ~u : (u | 0x80000000u);
  unsigned int skey = ~mfl;
  keys[tid] = ((unsigned long long)skey << 32) | (unsigned int)p;
}

// ---- kernel 3: one bitonic compare-exchange step (per-batch segments) -----

__global__ void vkp_bitonic(unsigned long long* __restrict__ keys,
                            unsigned int j, unsigned int k) {
  unsigned int tid = blockIdx.x * 256 + threadIdx.x;   // 8*65536 threads
  unsigned int b = tid >> 16, i = tid & (NP - 1);
  unsigned int l = i ^ j;
  if (l > i) {
    unsigned long long* base = keys + ((size_t)b << 16);
    unsigned long long a = base[i], c = base[l];
    bool asc = ((i & k) == 0);
    if ((a > c) == asc) { base[i] = c; base[l] = a; }
  }
}

// ---- kernel 4: per-pixel inverse channel norm via WMMA Gram diagonal ------
// One wave handles 16 pixels x 64 channels. A (16x4 f32) and B (4x16 f32)
// tiles coincide lane-for-lane (Gram is symmetric), so a single float2/lane
// feeds both operands of v_wmma_f32_16x16x4_f32; 16 chained WMMAs accumulate
// D = A A^T whose diagonal is sum_c d[p,c]^2.

__global__ void vkp_invnorm(const float* __restrict__ desc,
                            float* __restrict__ invn) {
  int wv = blockIdx.x * 8 + (threadIdx.x >> 5);   // [0, 32768)
  int lane = threadIdx.x & 31;
  int pixbase = wv << 4;                          // global pixel slot (b folded)
  int b = pixbase >> 16;
  int p = pixbase & (NP - 1);
  int half = lane >> 4;
  int pr = lane & 15;
  const float* dbase = desc + ((size_t)b << 22);  // b*64*65536

#if __has_builtin(__builtin_amdgcn_wmma_f32_16x16x4_f32)
  v8f c = {};
#pragma unroll
  for (int kb = 0; kb < CC; kb += 4) {
    int ch = kb + (half ? 2 : 0);                 // A layout: K={0,1}|{2,3} per half
    v2f a;
    a.x = dbase[((size_t)ch << 16) + p + pr];
    a.y = dbase[((size_t)(ch + 1) << 16) + p + pr];
    c = __builtin_amdgcn_wmma_f32_16x16x4_f32(false, a, false, a,
                                              (short)0, c, false, false);
  }
  // C/D layout: lane L<16 holds (M=v, N=L); lane L>=16 holds (M=v+8, N=L-16).
  // Diagonal (p,p): p<8 -> lane p, elem p;  p>=8 -> lane p+16, elem p-8.
  // Select via v_cndmask chain (sel = lane&7 matches both owner lane groups);
  // avoids the scratch spill a dynamic vector index would generate.
  int sel = lane & 7;
  float dg = c[0];
  dg = (sel == 1) ? c[1] : dg;
  dg = (sel == 2) ? c[2] : dg;
  dg = (sel == 3) ? c[3] : dg;
  dg = (sel == 4) ? c[4] : dg;
  dg = (sel == 5) ? c[5] : dg;
  dg = (sel == 6) ? c[6] : dg;
  dg = (sel == 7) ? c[7] : dg;
  float inv = 1.0f / fmaxf(sqrtf(dg), 1e-12f);
  if (lane < 8) {
    invn[pixbase + lane] = inv;
  } else if (lane >= 24) {
    invn[pixbase + lane - 16] = inv;
  }
#else
  // VALU fallback: lane pr sums 4 channels' squares, shfl-reduce across halves.
  float ss0 = 0.f, ss1 = 0.f;
  for (int kb = 0; kb < CC; kb += 4) {
    int ch = kb + (half ? 2 : 0);
    float a0 = dbase[((size_t)ch << 16) + p + pr];
    float a1 = dbase[((size_t)(ch + 1) << 16) + p + pr];
    ss0 += a0 * a0; ss1 += a1 * a1;
  }
  float ss = ss0 + ss1;
  ss += __shfl_xor(ss, 16, 32);
  if (lane < 16) invn[pixbase + lane] = 1.0f / fmaxf(sqrtf(ss), 1e-12f);
#endif
}

// ---- kernel 5: emit mk/scores + bicubic feats (one wave per keypoint) -----

__global__ void vkp_feats(const float* __restrict__ desc,
                          const float* __restrict__ invn,
                          const unsigned long long* __restrict__ keys,
                          float* __restrict__ out_mk,
                          float* __restrict__ out_sc,
                          float* __restrict__ out_ft) {
  int wv = blockIdx.x * 8 + (threadIdx.x >> 5);   // [0, 8*4096)
  int lane = threadIdx.x & 31;
  int b = wv >> 12;
  int kk = wv & (TOPK - 1);

  unsigned long long key = keys[((size_t)b << 16) + kk];
  unsigned int skey = (unsigned int)(key >> 32);
  unsigned int mfl = ~skey;
  unsigned int u = (mfl & 0x80000000u) ? (mfl ^ 0x80000000u) : ~mfl;
  float score = __uint_as_float(u);
  unsigned int pix = (unsigned int)(key & 0xFFFFFFFFull);

  int px, py;
  if (score == -1.0f) { px = 0; py = 0; }         // padding / kp==(0,0): mk=(0,0)
  else { px = (int)(pix & 255u); py = (int)(pix >> 8); }

  int oi = b * TOPK + kk;
  if (lane == 0) {
    out_mk[2 * oi]     = 4.0f * (float)px;
    out_mk[2 * oi + 1] = 4.0f * (float)py;
    out_sc[oi] = score;
  }

  float ix = unnorm_q(px), iy = unnorm_q(py);
  float fx0 = floorf(ix), fy0 = floorf(iy);
  int x0 = (int)fx0, y0 = (int)fy0;
  float wx[4], wy[4];
  cubw(ix - fx0, wx);
  cubw(iy - fy0, wy);

  int c0 = lane, c1 = lane + 32;
  const float* d0 = desc + (((size_t)b * CC + c0) << 16);
  const float* d1 = desc + (((size_t)b * CC + c1) << 16);
  const float* iv = invn + ((size_t)b << 16);

  float a0 = 0.f, a1 = 0.f;
#pragma unroll
  for (int j = 0; j < 4; j++) {
    int yy = y0 + j - 1;
    float r0 = 0.f, r1 = 0.f;
    if (yy >= 0 && yy < Hm) {
#pragma unroll
      for (int i = 0; i < 4; i++) {
        int xx = x0 + i - 1;
        if (xx >= 0 && xx < Wm) {
          int o = yy * Wm + xx;
          float s = iv[o];                 // channel-normalized desc tap
          r0 += wx[i] * (d0[o] * s);
          r1 += wx[i] * (d1[o] * s);
        }
      }
    }
    a0 += wy[j] * r0;
    a1 += wy[j] * r1;
  }

  float ss = a0 * a0 + a1 * a1;
#pragma unroll
  for (int off = 16; off > 0; off >>= 1) ss += __shfl_xor(ss, off, 32);
  float rn = 1.0f / fmaxf(sqrtf(ss), 1e-12f);

  float* fo = out_ft + ((size_t)oi << 6);  // [B,K,C]
  fo[c0] = a0 * rn;
  fo[c1] = a1 * rn;
}

// ---------------------------------------------------------------------------

extern "C" void kernel_launch(void* const* d_in, const int* in_sizes, int n_in,
                              void* d_out, int out_size, void* d_ws, size_t ws_size,
                              hipStream_t stream) {
  const float* rel  = (const float*)d_in[0];
  const float* var  = (const float*)d_in[1];
  const float* desc = (const float*)d_in[2];
  (void)in_sizes; (void)n_in; (void)out_size; (void)ws_size;

  char* ws = (char*)d_ws;
  unsigned int* mm = (unsigned int*)ws;                                 // 2 u32
  float* svar = (float*)(ws + 256);                                     // 2 MB
  float* invn = svar + (size_t)Bv * NP;                                 // 2 MB
  unsigned long long* keys =
      (unsigned long long*)(ws + 256 + 2 * (size_t)Bv * NP * 4);        // 4 MB

  float* out_mk = (float*)d_out;                  // [8,4096,2]
  float* out_sc = out_mk + Bv * TOPK * 2;         // [8,4096]
  float* out_ft = out_sc + Bv * TOPK;             // [8,4096,64]

  vkp_init<<<1, 1, 0, stream>>>(mm);
  vkp_nms<<<Bv * 16, 256, 0, stream>>>(rel, var, svar, mm);
  vkp_score<<<(Bv * NP) / 256, 256, 0, stream>>>(rel, svar, mm, keys);
  for (unsigned int k = 2; k <= NP; k <<= 1)
    for (unsigned int j = k >> 1; j > 0; j >>= 1)
      vkp_bitonic<<<(Bv * NP) / 256, 256, 0, stream>>>(keys, j, k);
  vkp_invnorm<<<(Bv * NP / 16) / 8, 256, 0, stream>>>(desc, invn);
  vkp_feats<<<(Bv * TOPK) / 8, 256, 0, stream>>>(desc, invn, keys,
                                                 out_mk, out_sc, out_ft);
}